// MemoryBank_3539053052646
// MI455X (gfx1250) — compile-verified
//
#include <hip/hip_runtime.h>
#include <hip/hip_bf16.h>

#define B_Q   64
#define BANK  4096
#define DIM   256
#define IMGEL 65536   // 1*256*256 floats per image

typedef float v2f __attribute__((ext_vector_type(2)));
typedef float v8f __attribute__((ext_vector_type(8)));

// ---------------------------------------------------------------------------
// Kernel 0: zero the 64 packed (key|~idx) max slots
// ---------------------------------------------------------------------------
__global__ void mb_init_best(unsigned long long* __restrict__ best) {
    int t = threadIdx.x;
    if (t < B_Q) best[t] = 0ull;   // all real float keys encode > 0
}

// ---------------------------------------------------------------------------
// Kernel 1: L2-normalize queries (torch F.normalize semantics, eps=1e-12)
// one block per row, 256 threads, LDS tree reduction
// ---------------------------------------------------------------------------
__global__ void mb_normalize(const float* __restrict__ q,
                             float* __restrict__ qn) {
    __shared__ float sdata[DIM];
    int row = blockIdx.x;
    int t   = threadIdx.x;
    float v = q[row * DIM + t];
    sdata[t] = v * v;
    __syncthreads();
    for (int s = DIM / 2; s > 0; s >>= 1) {
        if (t < s) sdata[t] += sdata[t + s];
        __syncthreads();
    }
    float denom = fmaxf(sqrtf(sdata[0]), 1e-12f);
    qn[row * DIM + t] = v / denom;
}

// ---------------------------------------------------------------------------
// helpers: order-preserving float->uint key, 64-bit xor shuffle
// ---------------------------------------------------------------------------
__device__ __forceinline__ unsigned int float_key(float f) {
    unsigned int bits = __float_as_uint(f);
    return (bits & 0x80000000u) ? ~bits : (bits | 0x80000000u);
}
__device__ __forceinline__ unsigned long long shfl_xor_u64(unsigned long long v, int mask) {
    unsigned int lo = (unsigned int)v;
    unsigned int hi = (unsigned int)(v >> 32);
    lo = __shfl_xor(lo, mask, 32);
    hi = __shfl_xor(hi, mask, 32);
    return ((unsigned long long)hi << 32) | lo;
}

// ---------------------------------------------------------------------------
// Kernel 2: sim = qn @ features^T via V_WMMA_F32_16X16X4_F32, fused max/argmax
// One wave per 16x16 tile. 4 M-tiles * 256 N-tiles = 1024 waves.
//
// f32 16x16x4 layouts (ISA 7.12.2):
//   A (16x4,MxK):  lane l -> M = l&15; VGPR0 = K{0|2}, VGPR1 = K{1|3} by l>>4
//   B (4x16,KxN):  lane l -> N = l&15; same K striping
//   C/D (16x16):   lane l -> N = l&15; VGPR v -> M = v + 8*(l>>4)
// sim(m,n) = sum_k qn[m][k] * features[n][k]  => B[k][n] = features[n][k]
// ---------------------------------------------------------------------------
__global__ void mb_sim_wmma(const float* __restrict__ qn,
                            const float* __restrict__ feats,
                            unsigned long long* __restrict__ best) {
    int wave  = (int)((blockIdx.x * blockDim.x + threadIdx.x) >> 5);
    int lane  = (int)(threadIdx.x & 31);
    int ntile = wave & 255;      // 0..255
    int mtile = wave >> 8;       // 0..3
    int lo16  = lane & 15;
    int half  = lane >> 4;

    const float* arow = qn    + (size_t)(mtile * 16 + lo16) * DIM;
    const float* brow = feats + (size_t)(ntile * 16 + lo16) * DIM;

    v8f acc = {};
#pragma unroll 8
    for (int k = 0; k < DIM; k += 4) {
        v2f a = *(const v2f*)(arow + k + 2 * half);
        v2f b = *(const v2f*)(brow + k + 2 * half);
        acc = __builtin_amdgcn_wmma_f32_16x16x4_f32(
            /*neg_a=*/false, a, /*neg_b=*/false, b,
            /*c_mod=*/(short)0, acc, /*reuse_a=*/false, /*reuse_b=*/false);
    }

    unsigned int col = (unsigned int)(ntile * 16 + lo16);
#pragma unroll
    for (int v = 0; v < 8; ++v) {
        unsigned long long p =
            ((unsigned long long)float_key(acc[v]) << 32) |
            (unsigned long long)(0xFFFFFFFFu - col);   // max key, then min col
        // reduce across the 16 lanes that share this output row
        for (int d = 1; d < 16; d <<= 1) {
            unsigned long long o = shfl_xor_u64(p, d);
            p = (o > p) ? o : p;
        }
        if (lo16 == 0) {
            int row = mtile * 16 + v + 8 * half;
            atomicMax(&best[row], p);
        }
    }
}

// ---------------------------------------------------------------------------
// Kernel 3: decode packed max -> scores (d_out tail) and indices (workspace)
// ---------------------------------------------------------------------------
__global__ void mb_finalize(const unsigned long long* __restrict__ best,
                            float* __restrict__ scores,
                            int* __restrict__ idx) {
    int r = threadIdx.x;
    if (r < B_Q) {
        unsigned long long p = best[r];
        unsigned int key  = (unsigned int)(p >> 32);
        unsigned int bits = (key & 0x80000000u) ? (key & 0x7FFFFFFFu) : ~key;
        scores[r] = __uint_as_float(bits);
        idx[r]    = (int)(0xFFFFFFFFu - (unsigned int)(p & 0xFFFFFFFFull));
    }
}

// ---------------------------------------------------------------------------
// Kernel 4: gather retrieved images, float4-coalesced (bandwidth-bound part)
// grid (64 rows, 64 chunks) x 256 threads -> 16384 float4 per image
// ---------------------------------------------------------------------------
__global__ void mb_gather(const float* __restrict__ images,
                          const int* __restrict__ idx,
                          float* __restrict__ out) {
    int row = blockIdx.x;
    int e   = (int)(blockIdx.y * blockDim.x + threadIdx.x);   // 0..16383
    long long src = (long long)idx[row] * (IMGEL / 4);
    const float4* s = (const float4*)images + src + e;
    float4*       d = (float4*)out + (long long)row * (IMGEL / 4) + e;
    *d = *s;
}

// ---------------------------------------------------------------------------
extern "C" void kernel_launch(void* const* d_in, const int* in_sizes, int n_in,
                              void* d_out, int out_size, void* d_ws, size_t ws_size,
                              hipStream_t stream) {
    const float* query  = (const float*)d_in[0];   // (64, 256)
    const float* feats  = (const float*)d_in[1];   // (4096, 256), pre-normalized
    const float* images = (const float*)d_in[2];   // (4096, 1, 256, 256)

    float* out_imgs   = (float*)d_out;                       // 64*65536 floats
    float* out_scores = (float*)d_out + (size_t)B_Q * IMGEL; // 64 floats

    // workspace layout
    char* ws = (char*)d_ws;
    float*              qn   = (float*)ws;                             // 64KB
    unsigned long long* best = (unsigned long long*)(ws + B_Q * DIM * sizeof(float));
    int*                idx  = (int*)(ws + B_Q * DIM * sizeof(float) + B_Q * sizeof(unsigned long long));

    mb_init_best<<<1, 64, 0, stream>>>(best);
    mb_normalize<<<B_Q, DIM, 0, stream>>>(query, qn);
    // 1024 waves, 8 waves/block -> 128 blocks
    mb_sim_wmma<<<128, 256, 0, stream>>>(qn, feats, best);
    mb_finalize<<<1, 64, 0, stream>>>(best, out_scores, idx);
    mb_gather<<<dim3(B_Q, IMGEL / 4 / 256), 256, 0, stream>>>(images, idx, out_imgs);
}